// GCN_36481452212847
// MI455X (gfx1250) — compile-verified
//
#include <hip/hip_runtime.h>
#include <hip/hip_bf16.h>
#include <stdint.h>

// ---------------------------------------------------------------------------
// GCN on MI455X (gfx1250, wave32).
//   layer: h = (x * norm_out) @ W ; agg = scatter_add(h[src] -> dst) ;
//          out = agg * norm_in + b ; (relu)
// GEMM uses v_wmma_f32_16x16x4_f32 (full fp32 precision, matches reference).
// SpMM uses HW f32 atomics at L2 scope (whole 51MB feature matrix fits in
// the 192MB L2, so gather/scatter never touches HBM).
// ---------------------------------------------------------------------------

typedef float v2f __attribute__((ext_vector_type(2)));
typedef float v8f __attribute__((ext_vector_type(8)));

static __device__ __forceinline__ void atomic_add_f32(float* p, float v) {
  // agent-scope relaxed fetch-add -> global_atomic_add_f32 (no CAS loop)
  __hip_atomic_fetch_add(p, v, __ATOMIC_RELAXED, __HIP_MEMORY_SCOPE_AGENT);
}

// ------------------------------- utility -----------------------------------

__global__ void zero_f32(float* __restrict__ p, long n) {
  long i = (long)blockIdx.x * blockDim.x + threadIdx.x;
  if (i < n) p[i] = 0.0f;
}

__global__ void degree_kernel(const int* __restrict__ src, const int* __restrict__ dst,
                              float* __restrict__ dout, float* __restrict__ din, int E) {
  int i = blockIdx.x * blockDim.x + threadIdx.x;
  if (i < E) {
    atomic_add_f32(&dout[src[i]], 1.0f);
    atomic_add_f32(&din[dst[i]], 1.0f);
  }
}

// in-place: deg -> clip(deg,1)^-0.5   (covers norm_out and norm_in, contiguous)
__global__ void norm_kernel(float* __restrict__ deg, long n) {
  long i = (long)blockIdx.x * blockDim.x + threadIdx.x;
  if (i < n) deg[i] = rsqrtf(fmaxf(deg[i], 1.0f));
}

// ------------------------------ WMMA GEMM ----------------------------------
// H[16t .. 16t+15, :] = (X * norm) @ W     X: [N,128]  W: [128,FOUT] row-major
// One wave32 per 16-row tile; 8 waves / block; W staged in LDS.
template <int FOUT>
__global__ void __launch_bounds__(256)
gemm_norm_wmma(const float* __restrict__ X, const float* __restrict__ W,
               const float* __restrict__ norm, float* __restrict__ H, int ntiles) {
  constexpr int FIN = 128;
  constexpr int NT = FOUT / 16;

  __shared__ __align__(16) float Wlds[FIN * FOUT];
  {
    const float4* wg = (const float4*)W;
    float4* wl = (float4*)Wlds;
    for (int i = threadIdx.x; i < FIN * FOUT / 4; i += blockDim.x) wl[i] = wg[i];
  }
  __syncthreads();

  const int wave = threadIdx.x >> 5;         // 0..7
  const int lane = threadIdx.x & 31;
  const int tile = blockIdx.x * 8 + wave;    // wave-uniform
  if (tile >= ntiles) return;                // uniform branch: EXEC stays all-1s

  const int half = lane >> 4;                // 0: lanes 0-15, 1: lanes 16-31
  const int lm   = lane & 15;                // M for A / N within tile for B,C,D
  const int row0 = tile * 16;

  const int arow = row0 + lm;                // A: both halves carry M = lm
  const float nrm = norm[arow];
  const float* __restrict__ xrow = X + (size_t)arow * FIN;

  v8f acc[NT];
#pragma unroll
  for (int t = 0; t < NT; ++t) acc[t] = (v8f)0.0f;

  for (int kk = 0; kk < FIN; kk += 4) {
    const int ka = kk + half * 2;            // ISA layout: low lanes K=kk..kk+1,
    v2f a;                                   //             high lanes K=kk+2..kk+3
    a.x = xrow[ka] * nrm;
    a.y = xrow[ka + 1] * nrm;
#pragma unroll
    for (int nt = 0; nt < NT; ++nt) {
      const int col = nt * 16 + lm;
      v2f b;                                 // B: lane = column, K striped like A
      b.x = Wlds[ka * FOUT + col];
      b.y = Wlds[(ka + 1) * FOUT + col];
      acc[nt] = __builtin_amdgcn_wmma_f32_16x16x4_f32(
          /*neg_a=*/false, a, /*neg_b=*/false, b,
          /*c_mod=*/(short)0, acc[nt], /*reuse_a=*/false, /*reuse_b=*/false);
    }
  }

  // D layout: VGPR v, lanes 0-15 -> M=v, lanes 16-31 -> M=8+v ; lane%16 = N
#pragma unroll
  for (int nt = 0; nt < NT; ++nt) {
    const int col = nt * 16 + lm;
    float* __restrict__ hp = H + (size_t)(row0 + half * 8) * FOUT + col;
#pragma unroll
    for (int r = 0; r < 8; ++r) hp[(size_t)r * FOUT] = acc[nt][r];
  }
}

// scalar fallback for a ragged tail (<16 rows); not used when N % 16 == 0
template <int FOUT>
__global__ void gemm_norm_scalar(const float* __restrict__ X, const float* __restrict__ W,
                                 const float* __restrict__ norm, float* __restrict__ H,
                                 int row_begin, int nrows) {
  constexpr int FIN = 128;
  int i = blockIdx.x * blockDim.x + threadIdx.x;
  if (i >= nrows * FOUT) return;
  int row = row_begin + i / FOUT;
  int col = i % FOUT;
  const float* xr = X + (size_t)row * FIN;
  float nrm = norm[row];
  float s = 0.0f;
  for (int k = 0; k < FIN; ++k) s += xr[k] * W[k * FOUT + col];
  H[(size_t)row * FOUT + col] = s * nrm;
}

// ------------------------------ SpMM scatter -------------------------------
// agg[dst[e], :] += h[src[e], :]   — one wave per edge, grid-stride.
template <int F>
__global__ void __launch_bounds__(256)
scatter_add(const float* __restrict__ H, const int* __restrict__ src,
            const int* __restrict__ dst, float* __restrict__ agg, int E) {
  const int lane = threadIdx.x & 31;
  int w = (blockIdx.x * blockDim.x + threadIdx.x) >> 5;
  const int nw = (gridDim.x * blockDim.x) >> 5;
  for (int e = w; e < E; e += nw) {
    const int s = src[e];
    const int d = dst[e];
    const float* __restrict__ hs = H + (size_t)s * F;
    float* __restrict__ ad = agg + (size_t)d * F;
    const int en = e + nw;
    if (en < E) {  // prefetch next edge's source row into L1/L2
      __builtin_prefetch(H + (size_t)src[en] * F + lane * 4, 0, 1);
    }
#pragma unroll
    for (int c = lane; c < F; c += 32) atomic_add_f32(&ad[c], hs[c]);
  }
}

// ------------------------------ finalize -----------------------------------
template <int F, bool RELU>
__global__ void finalize_kernel(const float* __restrict__ agg, const float* __restrict__ norm_in,
                                const float* __restrict__ bias, float* __restrict__ out, long n_elems) {
  constexpr int LOG = (F == 128) ? 7 : 6;
  long i = (long)blockIdx.x * blockDim.x + threadIdx.x;
  if (i >= n_elems) return;
  int row = (int)(i >> LOG);
  int col = (int)(i & (F - 1));
  float v = agg[i] * norm_in[row] + bias[col];
  if (RELU) v = fmaxf(v, 0.0f);
  out[i] = v;
}

// ------------------------------ launcher -----------------------------------

extern "C" void kernel_launch(void* const* d_in, const int* in_sizes, int n_in,
                              void* d_out, int out_size, void* d_ws, size_t ws_size,
                              hipStream_t stream) {
  (void)n_in; (void)out_size; (void)ws_size;
  const float* in_feat = (const float*)d_in[0];
  const int*   src     = (const int*)d_in[1];
  const int*   dst     = (const int*)d_in[2];
  const float* W1 = (const float*)d_in[3];
  const float* b1 = (const float*)d_in[4];
  const float* W2 = (const float*)d_in[5];
  const float* b2 = (const float*)d_in[6];
  const float* W3 = (const float*)d_in[7];
  const float* b3 = (const float*)d_in[8];

  const int N = in_sizes[0] / 128;
  const int E = in_sizes[1];

  float* ws       = (float*)d_ws;
  float* norm_out = ws;                       // N   (deg_out -> norm in place)
  float* norm_in  = ws + N;                   // N
  float* h        = ws + 2 * (size_t)N;       // N*128
  float* agg      = h + (size_t)N * 128;      // N*128
  float* xbuf     = agg + (size_t)N * 128;    // N*128
  float* out      = (float*)d_out;            // N*64

  const int T = 256;
  const long twoN = 2L * N;

  // ---- degrees & norms ----
  zero_f32<<<dim3((unsigned)((twoN + T - 1) / T)), dim3(T), 0, stream>>>(norm_out, twoN);
  degree_kernel<<<dim3((unsigned)((E + T - 1) / T)), dim3(T), 0, stream>>>(src, dst, norm_out, norm_in, E);
  norm_kernel<<<dim3((unsigned)((twoN + T - 1) / T)), dim3(T), 0, stream>>>(norm_out, twoN);

  const int ntiles = N / 16;
  const int rem = N - ntiles * 16;
  const unsigned gemm_blocks = (unsigned)((ntiles + 7) / 8);
  const unsigned scat_blocks = 8192;

  // ---- layer 1: 128 -> 128, relu ----
  {
    const long ne = (long)N * 128;
    if (ntiles)
      gemm_norm_wmma<128><<<dim3(gemm_blocks), dim3(T), 0, stream>>>(in_feat, W1, norm_out, h, ntiles);
    if (rem)
      gemm_norm_scalar<128><<<dim3((unsigned)((rem * 128 + T - 1) / T)), dim3(T), 0, stream>>>(
          in_feat, W1, norm_out, h, ntiles * 16, rem);
    zero_f32<<<dim3((unsigned)((ne + T - 1) / T)), dim3(T), 0, stream>>>(agg, ne);
    scatter_add<128><<<dim3(scat_blocks), dim3(T), 0, stream>>>(h, src, dst, agg, E);
    finalize_kernel<128, true><<<dim3((unsigned)((ne + T - 1) / T)), dim3(T), 0, stream>>>(
        agg, norm_in, b1, xbuf, ne);
  }

  // ---- layer 2: 128 -> 128, relu ----
  {
    const long ne = (long)N * 128;
    if (ntiles)
      gemm_norm_wmma<128><<<dim3(gemm_blocks), dim3(T), 0, stream>>>(xbuf, W2, norm_out, h, ntiles);
    if (rem)
      gemm_norm_scalar<128><<<dim3((unsigned)((rem * 128 + T - 1) / T)), dim3(T), 0, stream>>>(
          xbuf, W2, norm_out, h, ntiles * 16, rem);
    zero_f32<<<dim3((unsigned)((ne + T - 1) / T)), dim3(T), 0, stream>>>(agg, ne);
    scatter_add<128><<<dim3(scat_blocks), dim3(T), 0, stream>>>(h, src, dst, agg, E);
    finalize_kernel<128, true><<<dim3((unsigned)((ne + T - 1) / T)), dim3(T), 0, stream>>>(
        agg, norm_in, b2, xbuf, ne);
  }

  // ---- layer 3: 128 -> 64, no relu, write d_out ----
  {
    const long ne = (long)N * 64;
    if (ntiles)
      gemm_norm_wmma<64><<<dim3(gemm_blocks), dim3(T), 0, stream>>>(xbuf, W3, norm_out, h, ntiles);
    if (rem)
      gemm_norm_scalar<64><<<dim3((unsigned)((rem * 64 + T - 1) / T)), dim3(T), 0, stream>>>(
          xbuf, W3, norm_out, h, ntiles * 16, rem);
    zero_f32<<<dim3((unsigned)((ne + T - 1) / T)), dim3(T), 0, stream>>>(agg, ne);
    scatter_add<64><<<dim3(scat_blocks), dim3(T), 0, stream>>>(h, src, dst, agg, E);
    finalize_kernel<64, false><<<dim3((unsigned)((ne + T - 1) / T)), dim3(T), 0, stream>>>(
        agg, norm_in, b3, out, ne);
  }
}